// loss_soft_add_5935644803607
// MI455X (gfx1250) — compile-verified
//
#include <hip/hip_runtime.h>
#include <stdint.h>

// ---------------- problem constants ----------------
#define N_COLS      25
#define ROWS        256                    // rows per tile (== threads per block)
#define TILE_FLOATS (ROWS * N_COLS)        // 6400 floats per array per tile
#define TILE_VEC4   (TILE_FLOATS / 4)      // 1600 x 16B async copies
#define THREADS     256
#define NBLK        3125                   // 15625 tiles / 3125 = exactly 5 tiles/block
#define NTILES      15625                  // 4,000,000 / 256

typedef float v2f __attribute__((ext_vector_type(2)));
typedef float v8f __attribute__((ext_vector_type(8)));

// CDNA5 async global->LDS copy (VGLOBAL encoding, ASYNCcnt-tracked).
// vdst = wave-relative LDS byte address (low 32 bits of the flat shared addr),
// vaddr = 64-bit global address, no saddr ("off" form).
__device__ __forceinline__ void async_copy_b128(uint32_t lds_byte_addr,
                                                const float* gptr) {
  asm volatile("global_load_async_to_lds_b128 %0, %1, off"
               :: "v"(lds_byte_addr), "v"(gptr)
               : "memory");
}

__device__ __forceinline__ void wait_async_zero() {
  asm volatile("s_wait_asynccnt 0x0" ::: "memory");
}

// Exact wave32 sum via V_WMMA_F32_16X16X4_F32.
// A (16x4 f32, documented layout): lane L<16 -> A[L][0]=s, A[L][1]=0;
// lane L>=16 -> A[L-16][2]=s, A[L-16][3]=0.  B = all ones (both VGPRs), so
// D[m][n] = sum_k A[m][k] = s_m + s_{m+16} regardless of B striping.
// D layout: VGPR r, lanes 0-15 hold M=r, lanes 16-31 hold M=r+8.
__device__ __forceinline__ float wave_sum_wmma(float s) {
  v2f a; a[0] = s;    a[1] = 0.0f;
  v2f b; b[0] = 1.0f; b[1] = 1.0f;
  v8f c = {};
  v8f d = __builtin_amdgcn_wmma_f32_16x16x4_f32(
      /*neg_a=*/false, a, /*neg_b=*/false, b,
      /*c_mod=*/(short)0, c, /*reuse_a=*/false, /*reuse_b=*/false);
  float t = ((d[0] + d[1]) + (d[2] + d[3])) + ((d[4] + d[5]) + (d[6] + d[7]));
  // lanes 0-15 now hold sum(s_0..7, s_16..23); lanes 16-31 hold the rest.
  t += __shfl_xor(t, 16, 32);
  return t;  // full wave sum in every lane
}

__global__ __launch_bounds__(THREADS)
void loss_partial_kernel(const float* __restrict__ pre,
                         const float* __restrict__ tar,
                         float* __restrict__ ws) {
  __shared__ float s_pre[TILE_FLOATS];   // 25,600 B
  __shared__ float s_tar[TILE_FLOATS];   // 25,600 B
  __shared__ float s_red[THREADS / 32];

  const int tid = threadIdx.x;
  float acc = 0.0f;

  // Exactly 5 tiles per block: perfectly balanced persistent-block stream.
  for (int tile = blockIdx.x; tile < NTILES; tile += NBLK) {
    const size_t base = (size_t)tile * TILE_FLOATS;

    // Stage tile into LDS: contiguous, 512B per async instruction per wave.
    for (int i = tid; i < TILE_VEC4; i += THREADS) {
      const size_t off = 4 * (size_t)i;
      async_copy_b128((uint32_t)(uintptr_t)&s_pre[4 * i], pre + base + off);
      async_copy_b128((uint32_t)(uintptr_t)&s_tar[4 * i], tar + base + off);
    }
    wait_async_zero();
    __syncthreads();

    // One row per thread. Stride-25 word access -> 32 distinct LDS banks.
    const float* p = &s_pre[tid * N_COLS];
    const float* t = &s_tar[tid * N_COLS];
    float r = 0.0f;
#pragma unroll
    for (int c = 0; c < 6; ++c) {
      const float a0 = p[4 * c + 0], a1 = p[4 * c + 1];
      const float a2 = p[4 * c + 2], a3 = p[4 * c + 3];
      const float m  = fmaxf(fmaxf(a0, a1), fmaxf(a2, a3));
      const float e0 = __expf(a0 - m), e1 = __expf(a1 - m);
      const float e2 = __expf(a2 - m), e3 = __expf(a3 - m);
      const float k  = 3.0f / (e0 + e1 + e2 + e3);   // 3*EXPAND*softmax
      r += fabsf(e0 * k - t[4 * c + 0]) + fabsf(e1 * k - t[4 * c + 1]) +
           fabsf(e2 * k - t[4 * c + 2]) + fabsf(e3 * k - t[4 * c + 3]);
    }
    acc += r * (1.0f / 24.0f);   // mean over 24 elements

    __syncthreads();  // all reads done before next tile's async stores land
  }

  // Block reduction: WMMA wave sum, then 8 wave partials in fixed order.
  const float w = wave_sum_wmma(acc);
  if ((tid & 31) == 0) s_red[tid >> 5] = w;
  __syncthreads();
  if (tid == 0) {
    float bsum = 0.0f;
#pragma unroll
    for (int i = 0; i < THREADS / 32; ++i) bsum += s_red[i];
    ws[blockIdx.x] = bsum;
  }
}

// Deterministic final reduction of NBLK partials in double precision.
__global__ __launch_bounds__(THREADS)
void loss_final_kernel(const float* __restrict__ ws, float* __restrict__ out,
                       int n) {
  __shared__ double red[THREADS];
  double acc = 0.0;
  for (int i = threadIdx.x; i < n; i += THREADS) acc += (double)ws[i];
  red[threadIdx.x] = acc;
  __syncthreads();
  for (int off = THREADS / 2; off > 0; off >>= 1) {
    if (threadIdx.x < off) red[threadIdx.x] += red[threadIdx.x + off];
    __syncthreads();
  }
  if (threadIdx.x == 0) out[0] = (float)red[0];
}

extern "C" void kernel_launch(void* const* d_in, const int* in_sizes, int n_in,
                              void* d_out, int out_size, void* d_ws,
                              size_t ws_size, hipStream_t stream) {
  const float* pre = (const float*)d_in[0];
  const float* tar = (const float*)d_in[1];
  float* ws = (float*)d_ws;          // needs NBLK*4 = 12.5 KB scratch
  float* out = (float*)d_out;

  loss_partial_kernel<<<NBLK, THREADS, 0, stream>>>(pre, tar, ws);
  loss_final_kernel<<<1, THREADS, 0, stream>>>(ws, out, NBLK);
}